// KNN_62629213110352
// MI455X (gfx1250) — compile-verified
//
#include <hip/hip_runtime.h>

// KNN (SelectKnn-style): per 4096-point segment, self + 64 nearest neighbours
// in squared L2 (D=4), radius^2 = 1.0 masking. Distances via
// V_WMMA_F32_16X16X4_F32 (M=16,N=16,K=4 == D), selection via LDS insertion
// lists with radius pruning. 64-wide candidate strips: 8 WMMAs per wave per
// barrier pair, float4 LDS reads in the selection scan, global prefetch of
// the next strip.

#define K1     65     // K + 1 (self)
#define QB     64     // queries per block (= blockDim.x, 2 waves)
#define CTILE  64     // candidates per iteration (4 WMMA col-subtiles)

typedef __attribute__((ext_vector_type(2))) float v2f;
typedef __attribute__((ext_vector_type(8))) float v8f;

__global__ __launch_bounds__(QB) void knn_wmma_kernel(
    const float4* __restrict__ coords,  // [B*S] float4
    int*   __restrict__ oidx,           // [B*S][K1]
    float* __restrict__ odist,          // [B*S][K1]
    int S)                              // points per segment (4096, pow2)
{
    __shared__ float distL[K1 * QB];     // sorted candidate distances per query
    __shared__ int   idxL [K1 * QB];     // matching global indices
    __shared__ float d2buf[QB * CTILE];  // 64x64 distance strip
    __shared__ float qn   [QB];          // query squared norms

    const int tid   = threadIdx.x;
    const int lane  = tid & 31;
    const int wave  = tid >> 5;
    const int seg   = blockIdx.y;
    const int qblk  = blockIdx.x * QB;   // segment-local base query row
    const int segbase = seg * S;         // global base row of this segment
    const int qs    = qblk + tid;        // segment-local query index
    const int qg    = segbase + qs;      // global query index

    // Own query norm -> LDS
    float4 myq = coords[qg];
    qn[tid] = myq.x*myq.x + myq.y*myq.y + myq.z*myq.z + myq.w*myq.w;

    // Seed list with self (sentinel -1 so it sorts first, like reference diag=-1)
    distL[0 * QB + tid] = -1.0f;
    idxL [0 * QB + tid] = qg;
    int cnt = 1;

    __syncthreads();

    // --- Persistent A fragments: 2 row-tiles of 16 queries per wave ---
    // 32-bit A 16x4 layout: lanes 0-15 = rows M with K=0 (v0) / K=1 (v1);
    // lanes 16-31 = same rows, K=2 / K=3.
    const int m  = lane & 15;
    const int kh = lane >> 4;
    v2f   afrag[2];
    float qnr[2][8];                     // query norms for this lane's C rows
    #pragma unroll
    for (int t = 0; t < 2; ++t) {
        const int rt = wave * 2 + t;     // row-tile 0..3
        float4 q = coords[segbase + qblk + rt * 16 + m];
        afrag[t].x = kh ? q.z : q.x;
        afrag[t].y = kh ? q.w : q.y;
        #pragma unroll
        for (int r = 0; r < 8; ++r)
            qnr[t][r] = qn[rt * 16 + r + kh * 8];
    }

    const float r2 = 1.0f;               // RADIUS^2

    const int nstrips = S / CTILE;       // 64
    for (int ct = 0; ct < nstrips; ++ct) {
        const int cbase = ct * CTILE;

        // Prefetch next strip (64 float4 = 1KB) while we compute this one.
        if (ct + 1 < nstrips)
            __builtin_prefetch(&coords[segbase + cbase + CTILE + tid], 0, 0);

        // 4 col-subtiles of 16 candidates -> 8 WMMAs per wave per strip
        #pragma unroll
        for (int sub = 0; sub < 4; ++sub) {
            // B fragment (4x16, mirror of A layout: lane = column N)
            float4 c = coords[segbase + cbase + sub * 16 + m];
            v2f bfrag;
            bfrag.x = kh ? c.z : c.x;
            bfrag.y = kh ? c.w : c.y;
            const float cn = c.x*c.x + c.y*c.y + c.z*c.z + c.w*c.w;

            #pragma unroll
            for (int t = 0; t < 2; ++t) {
                v8f acc = {};
                acc = __builtin_amdgcn_wmma_f32_16x16x4_f32(
                    false, afrag[t], false, bfrag, (short)0, acc, false, false);
                const int rt = wave * 2 + t;
                #pragma unroll
                for (int r = 0; r < 8; ++r) {
                    const int M = rt * 16 + r + kh * 8;   // block-local query row
                    float d2 = fmaxf(qnr[t][r] + cn - 2.0f * acc[r], 0.0f);
                    d2buf[M * CTILE + sub * 16 + m] = d2;
                }
            }
        }
        __syncthreads();

        // Selection: each thread scans its query's 64 candidates (float4 LDS reads)
        const float4* rowp = (const float4*)&d2buf[tid * CTILE];
        #pragma unroll 1
        for (int v = 0; v < CTILE / 4; ++v) {
            const float4 dv = rowp[v];
            const float dvals[4] = { dv.x, dv.y, dv.z, dv.w };
            #pragma unroll
            for (int e = 0; e < 4; ++e) {
                const int jl = cbase + v * 4 + e;   // segment-local candidate
                if (jl == qs) continue;             // self handled by seed
                const float d = dvals[e];
                if (d > r2) continue;               // radius prune
                int pos;
                if (cnt == K1) {
                    if (d >= distL[(K1 - 1) * QB + tid]) continue;
                    pos = K1 - 1;                   // replace current worst
                } else {
                    pos = cnt++;
                }
                while (pos > 0 && distL[(pos - 1) * QB + tid] > d) {
                    distL[pos * QB + tid] = distL[(pos - 1) * QB + tid];
                    idxL [pos * QB + tid] = idxL [(pos - 1) * QB + tid];
                    --pos;
                }
                distL[pos * QB + tid] = d;
                idxL [pos * QB + tid] = segbase + jl;
            }
        }
        __syncthreads();
    }

    // Emit: listed entries (all <= r2), rest -> idx=-1, dist=0. Self -1 -> 0.
    for (int e = 0; e < K1; ++e) {
        int   oi;
        float od;
        if (e < cnt) {
            od = fmaxf(distL[e * QB + tid], 0.0f);
            oi = idxL [e * QB + tid];
        } else {
            od = 0.0f;
            oi = -1;
        }
        oidx [(size_t)qg * K1 + e] = oi;
        odist[(size_t)qg * K1 + e] = od;
    }
}

extern "C" void kernel_launch(void* const* d_in, const int* in_sizes, int n_in,
                              void* d_out, int out_size, void* d_ws, size_t ws_size,
                              hipStream_t stream) {
    const float4* coords = (const float4*)d_in[0];
    // d_in[1] = row_splits (equal-size segments; geometry derived from sizes)
    const int B = in_sizes[1] - 1;          // 12
    const int N = in_sizes[0] / 4;          // total points (B*S)
    const int S = N / B;                    // 4096 points per segment

    int*   oidx  = (int*)d_out;                       // first N*K1 ints
    float* odist = (float*)d_out + (size_t)N * K1;    // then N*K1 floats

    dim3 grid(S / QB, B);
    knn_wmma_kernel<<<grid, QB, 0, stream>>>(coords, oidx, odist, S);
}